// Pointnet2_g_12017318494784
// MI455X (gfx1250) — compile-verified
//
#include <hip/hip_runtime.h>
#include <cstdint>
#include <cstddef>

typedef __attribute__((ext_vector_type(2))) float v2f;
typedef __attribute__((ext_vector_type(8))) float v8f;

#define DEV __device__ __forceinline__

// ---------------------------------------------------------------------------
// WMMA helpers (V_WMMA_F32_16X16X4_F32, fp32 matrix core path)
// ---------------------------------------------------------------------------
DEV v8f wmma4(v2f a, v2f b, v8f c) {
  return __builtin_amdgcn_wmma_f32_16x16x4_f32(false, a, false, b, (short)0, c,
                                               false, false);
}

// A fragment: 16x4 tile of X (row-major, LDS), rows m0..m0+15, cols k0..k0+3.
// lanes 0-15: K=k0,k0+1 ; lanes 16-31: K=k0+2,k0+3 (ISA 32-bit A layout)
DEV v2f frag_a(const float* X, int lane, int m0, int k0, int ld) {
  int half = lane >> 4, l = lane & 15;
  const float* p = X + (size_t)(m0 + l) * ld + k0 + half * 2;
  v2f a;
  a.x = p[0];
  a.y = p[1];
  return a;
}

// B fragment from pair-packed, zero-padded weights:
// wtp[(k/2)*Cout + n][2] = { Wt[k][n], Wt[k+1][n] }  -> one coalesced b64/lane.
DEV v2f frag_bp(const float* wtp, int cout, int n0, int k0, int lane) {
  int half = lane >> 4, l = lane & 15;
  int pair = (k0 >> 1) + half;  // k0 multiple of 4
  return *((const v2f*)wtp + (size_t)pair * cout + n0 + l);
}

// Accumulate one 16-col output tile (n0) over all K: a0 = rows 0-15,
// a1 = rows 16-31 (if ROWS==32). B-fragments hoisted in 16-step blocks and
// reused across both M-tiles.
template <int KPAD, int COUT, int ROWS>
DEV void layer_core(const float* X, const float* wtp, int n0, int lane, v8f& a0,
                    v8f& a1) {
  constexpr int KS = KPAD / 4;
  constexpr int KMAIN = (KS / 16) * 16;
  for (int s0 = 0; s0 < KMAIN; s0 += 16) {
    v2f bf[16];
#pragma unroll
    for (int s = 0; s < 16; ++s) bf[s] = frag_bp(wtp, COUT, n0, (s0 + s) * 4, lane);
#pragma unroll
    for (int s = 0; s < 16; ++s)
      a0 = wmma4(frag_a(X, lane, 0, (s0 + s) * 4, KPAD), bf[s], a0);
    if constexpr (ROWS == 32) {
#pragma unroll
      for (int s = 0; s < 16; ++s)
        a1 = wmma4(frag_a(X, lane, 16, (s0 + s) * 4, KPAD), bf[s], a1);
    }
  }
  if constexpr (KS % 16) {
    constexpr int KT = KS % 16;
    v2f bf[KT];
#pragma unroll
    for (int s = 0; s < KT; ++s)
      bf[s] = frag_bp(wtp, COUT, n0, (KMAIN + s) * 4, lane);
#pragma unroll
    for (int s = 0; s < KT; ++s)
      a0 = wmma4(frag_a(X, lane, 0, (KMAIN + s) * 4, KPAD), bf[s], a0);
    if constexpr (ROWS == 32) {
#pragma unroll
      for (int s = 0; s < KT; ++s)
        a1 = wmma4(frag_a(X, lane, 16, (KMAIN + s) * 4, KPAD), bf[s], a1);
    }
  }
}

// O = relu(X * Wt + bias); X: ROWS x KPAD (LDS), O: ROWS x COUT (LDS).
template <int KPAD, int COUT, int ROWS>
DEV void layer_lds(const float* X, float* O, const float* wtp, const float* bias,
                   int lane) {
  int col_l = lane & 15;
  int rhalf = (lane >> 4) << 3;
  for (int n0 = 0; n0 < COUT; n0 += 16) {
    v8f a0 = {}, a1 = {};
    layer_core<KPAD, COUT, ROWS>(X, wtp, n0, lane, a0, a1);
    float bv = bias[n0 + col_l];
#pragma unroll
    for (int v = 0; v < 8; ++v) {
      float x = a0[v] + bv;
      O[(size_t)(rhalf + v) * COUT + n0 + col_l] = x > 0.0f ? x : 0.0f;
    }
    if constexpr (ROWS == 32) {
#pragma unroll
      for (int v = 0; v < 8; ++v) {
        float x = a1[v] + bv;
        O[(size_t)(16 + rhalf + v) * COUT + n0 + col_l] = x > 0.0f ? x : 0.0f;
      }
    }
  }
}

// out[c] (or atomic-max merge) = max over rows of relu(X * Wt + b).
template <int KPAD, int COUT, int ROWS>
DEV void layer_max(const float* X, const float* wtp, const float* bias, int lane,
                   float* out, bool use_atomic) {
  int col_l = lane & 15;
  for (int n0 = 0; n0 < COUT; n0 += 16) {
    v8f a0 = {}, a1 = {};
    layer_core<KPAD, COUT, ROWS>(X, wtp, n0, lane, a0, a1);
    float bv = bias[n0 + col_l];
    float cm = 0.0f;  // relu outputs >= 0
#pragma unroll
    for (int v = 0; v < 8; ++v) {
      float x = a0[v] + bv;
      cm = fmaxf(cm, x > 0.0f ? x : 0.0f);
    }
    if constexpr (ROWS == 32) {
#pragma unroll
      for (int v = 0; v < 8; ++v) {
        float x = a1[v] + bv;
        cm = fmaxf(cm, x > 0.0f ? x : 0.0f);
      }
    }
    cm = fmaxf(cm, __shfl_xor(cm, 16, 32));
    if (lane < 16) {
      if (use_atomic)
        atomicMax((unsigned int*)(out + n0 + lane), __float_as_uint(cm));
      else
        out[n0 + lane] = cm;
    }
  }
}

// ---------------------------------------------------------------------------
// Weight transpose + zero-pad + pair-pack:
// wtp[((k/2)*cout + n)*2 + k%2] = (k < kreal) ? w[n][k] : 0
// ---------------------------------------------------------------------------
__global__ void wtr_kernel(const float* __restrict__ w, float* __restrict__ wtp,
                           int kreal, int kpad, int cout) {
  int idx = blockIdx.x * blockDim.x + threadIdx.x;
  if (idx < kpad * cout) {
    int k = idx / cout, n = idx - k * cout;
    float v = (k < kreal) ? w[(size_t)n * kreal + k] : 0.0f;
    wtp[((size_t)(k >> 1) * cout + n) * 2 + (k & 1)] = v;
  }
}

// Pair-pack node_fea ([B][512][64] -> [B][256][64][2]) for the cr GEMM.
__global__ void nfp_kernel(const float* __restrict__ nf,
                           float* __restrict__ nfp) {
  size_t i = (size_t)blockIdx.x * blockDim.x + threadIdx.x;
  if (i < (size_t)32 * 512 * 64) {
    int b = (int)(i >> 15);
    int r = (int)(i & 32767);
    int k = r >> 6, c = r & 63;
    nfp[((size_t)b << 15) + ((((size_t)(k >> 1) * 64) + c) << 1) + (k & 1)] =
        nf[i];
  }
}

// ---------------------------------------------------------------------------
// Layout / setup kernels
// ---------------------------------------------------------------------------
__global__ void transpose_kernel(const float* __restrict__ xyz,  // [B][3][2048]
                                 float* __restrict__ pts) {      // [B][2048][3]
  int b = blockIdx.y;
  int n = blockIdx.x * blockDim.x + threadIdx.x;
  if (n < 2048) {
    for (int c = 0; c < 3; ++c)
      pts[((size_t)b * 2048 + n) * 3 + c] = xyz[((size_t)b * 3 + c) * 2048 + n];
  }
}

__global__ void zero_kernel(float* __restrict__ xout,    // 32*1024
                            float* __restrict__ l2pts) { // 32*128*256
  size_t i = (size_t)blockIdx.x * blockDim.x + threadIdx.x;
  if (i < 32 * 1024) xout[i] = 0.0f;
  if (i < (size_t)32 * 128 * 256) l2pts[i] = 0.0f;
}

// ---------------------------------------------------------------------------
// Farthest point sampling: one block per batch, sequential over npoint.
// ---------------------------------------------------------------------------
__global__ __launch_bounds__(256) void fps_kernel(const float* __restrict__ pts,
                                                  int N, int npoint,
                                                  int* __restrict__ fidx) {
  __shared__ float pl[2048 * 3];
  __shared__ float dist[2048];
  __shared__ float redv[256];
  __shared__ int redi[256];
  int b = blockIdx.x;
  int t = threadIdx.x;
  const float* P = pts + (size_t)b * N * 3;
  for (int i = t; i < N; i += 256) {
    pl[i * 3 + 0] = P[i * 3 + 0];
    pl[i * 3 + 1] = P[i * 3 + 1];
    pl[i * 3 + 2] = P[i * 3 + 2];
    dist[i] = 1e10f;
  }
  __syncthreads();
  int far = 0;
  for (int it = 0; it < npoint; ++it) {
    if (t == 0) fidx[(size_t)b * npoint + it] = far;
    float cx = pl[far * 3 + 0], cy = pl[far * 3 + 1], cz = pl[far * 3 + 2];
    float bv = -1.0f;
    int bi = 0;
    for (int i = t; i < N; i += 256) {
      float dx = pl[i * 3 + 0] - cx;
      float dy = pl[i * 3 + 1] - cy;
      float dz = pl[i * 3 + 2] - cz;
      float d = dx * dx + dy * dy + dz * dz;
      float dd = fminf(dist[i], d);
      dist[i] = dd;
      if (dd > bv) { bv = dd; bi = i; }  // strided -> earliest index on ties
    }
    redv[t] = bv;
    redi[t] = bi;
    __syncthreads();
    for (int s = 128; s > 0; s >>= 1) {
      if (t < s) {
        float v2 = redv[t + s];
        int i2 = redi[t + s];
        if (v2 > redv[t] || (v2 == redv[t] && i2 < redi[t])) {
          redv[t] = v2;
          redi[t] = i2;
        }
      }
      __syncthreads();
    }
    far = redi[0];
    __syncthreads();
  }
}

__global__ void gather_xyz_kernel(const float* __restrict__ pts,
                                  const int* __restrict__ fidx,
                                  float* __restrict__ out, int N, int npoint) {
  int b = blockIdx.y;
  int i = blockIdx.x * blockDim.x + threadIdx.x;
  if (i < npoint) {
    int p = fidx[(size_t)b * npoint + i];
    for (int c = 0; c < 3; ++c)
      out[((size_t)b * npoint + i) * 3 + c] = pts[((size_t)b * N + p) * 3 + c];
  }
}

// ---------------------------------------------------------------------------
// Ball query: one wave per centroid; ballot + prefix-popcount compaction.
// ---------------------------------------------------------------------------
__global__ void ball_query_kernel(const float* __restrict__ pts,   // [B][N][3]
                                  const float* __restrict__ cent,  // [B][S][3]
                                  int N, int S, int nsample, float r2,
                                  int nGroups, int* __restrict__ gidx) {
  int gtid = blockIdx.x * blockDim.x + threadIdx.x;
  int wid = gtid >> 5;
  int lane = threadIdx.x & 31;
  if (wid >= nGroups) return;  // uniform per wave
  int b = wid / S;
  float cx = cent[(size_t)wid * 3 + 0];
  float cy = cent[(size_t)wid * 3 + 1];
  float cz = cent[(size_t)wid * 3 + 2];
  int* out = gidx + (size_t)wid * nsample;
  const float* P = pts + (size_t)b * N * 3;
  int cnt = 0;
  int firstIdx = 0;
  for (int n0 = 0; n0 < N && cnt < nsample; n0 += 32) {
    int n = n0 + lane;
    float dx = P[n * 3 + 0] - cx;
    float dy = P[n * 3 + 1] - cy;
    float dz = P[n * 3 + 2] - cz;
    float d = dx * dx + dy * dy + dz * dz;
    bool pred = d <= r2;
    unsigned mask = (unsigned)__ballot(pred);
    int pos = cnt + __popc(mask & ((1u << lane) - 1u));
    if (pred && pos < nsample) out[pos] = n;
    if (cnt == 0 && mask) firstIdx = n0 + __builtin_ctz(mask);
    cnt += __popc(mask);
  }
  if (cnt < nsample) {
    for (int i = cnt + lane; i < nsample; i += 32) out[i] = firstIdx;
  }
}

// ---------------------------------------------------------------------------
// SA1: one wave per (b,s) group. 32 samples x (3 -> 64 -> 64 -> 128), maxpool.
// ---------------------------------------------------------------------------
__global__ __launch_bounds__(32) void sa1_kernel(
    const float* __restrict__ pts,   // [B][2048][3]
    const float* __restrict__ cent,  // [B][512][3]
    const int* __restrict__ gidx,    // [B][512][32]
    const float* wt1, const float* b1, const float* wt2, const float* b2,
    const float* wt3, const float* b3, float* __restrict__ l1_pts,
    float* __restrict__ node_fea) {
  __shared__ float X0[32 * 4];
  __shared__ float X1[32 * 64];
  __shared__ float X2[32 * 64];
  int g = blockIdx.x;  // b*512+s
  int lane = threadIdx.x;
  int b = g >> 9;
  float cx = cent[(size_t)g * 3 + 0];
  float cy = cent[(size_t)g * 3 + 1];
  float cz = cent[(size_t)g * 3 + 2];
  int p = gidx[(size_t)g * 32 + lane];
  const float* pp = pts + ((size_t)b * 2048 + p) * 3;
  X0[lane * 4 + 0] = pp[0] - cx;
  X0[lane * 4 + 1] = pp[1] - cy;
  X0[lane * 4 + 2] = pp[2] - cz;
  X0[lane * 4 + 3] = 0.0f;
  __syncthreads();
  layer_lds<4, 64, 32>(X0, X1, wt1, b1, lane);
  __syncthreads();
  for (int c = lane; c < 64; c += 32) {  // node_fea = max over 32 rows of X1
    float m = 0.0f;
    for (int r = 0; r < 32; ++r) m = fmaxf(m, X1[r * 64 + c]);
    node_fea[(size_t)g * 64 + c] = m;
  }
  layer_lds<64, 64, 32>(X1, X2, wt2, b2, lane);
  __syncthreads();
  layer_max<64, 128, 32>(X2, wt3, b3, lane, l1_pts + (size_t)g * 128, false);
}

// ---------------------------------------------------------------------------
// SA2: one wave per (group, chunk-of-32-samples). 131 -> 128 -> 128 -> 256.
// ---------------------------------------------------------------------------
__global__ __launch_bounds__(32) void sa2_kernel(
    const float* __restrict__ l1_xyz,  // [B][512][3]
    const float* __restrict__ l1_pts,  // [B][512][128]
    const float* __restrict__ cent,    // [B][128][3]
    const int* __restrict__ gidx,      // [B][128][64]
    const float* wt1, const float* b1, const float* wt2, const float* b2,
    const float* wt3, const float* b3, float* __restrict__ l2_pts) {
  __shared__ float XA[32 * 132];
  __shared__ float XB[32 * 128];
  __shared__ float XC[32 * 128];
  int blk = blockIdx.x;
  int g = blk >> 1;     // b*128+s
  int chunk = blk & 1;  // which 32 of the 64 samples
  int lane = threadIdx.x;
  int b = g >> 7;
  int p = gidx[(size_t)g * 64 + chunk * 32 + lane];
  for (int j = 0; j < 32; ++j) {
    int pj = __shfl(p, j, 32);
    const float4* fp = (const float4*)(l1_pts + ((size_t)b * 512 + pj) * 128);
    float4 q = fp[lane];  // coalesced b128
    float* dst = XA + j * 132 + 3 + lane * 4;
    dst[0] = q.x; dst[1] = q.y; dst[2] = q.z; dst[3] = q.w;
    if (lane < 3)
      XA[j * 132 + lane] =
          l1_xyz[((size_t)b * 512 + pj) * 3 + lane] - cent[(size_t)g * 3 + lane];
    if (lane == 3) XA[j * 132 + 131] = 0.0f;
  }
  __syncthreads();
  layer_lds<132, 128, 32>(XA, XB, wt1, b1, lane);
  __syncthreads();
  layer_lds<128, 128, 32>(XB, XC, wt2, b2, lane);
  __syncthreads();
  layer_max<128, 256, 32>(XC, wt3, b3, lane, l2_pts + (size_t)g * 256, true);
}

// ---------------------------------------------------------------------------
// SA3 (group-all): one wave per (batch, chunk-of-16-rows). 259->256->512->1024.
// ---------------------------------------------------------------------------
__global__ __launch_bounds__(32) void sa3_kernel(
    const float* __restrict__ l2_xyz,  // [B][128][3]
    const float* __restrict__ l2_pts,  // [B][128][256]
    const float* wt1, const float* b1, const float* wt2, const float* b2,
    const float* wt3, const float* b3, float* __restrict__ xout) {  // [B][1024]
  __shared__ float XA[16 * 512];  // Xin(16x260), then X2(16x512)
  __shared__ float XB[16 * 256];  // X1
  int blk = blockIdx.x;
  int b = blk >> 3;
  int chunk = blk & 7;
  int lane = threadIdx.x;
  for (int j = 0; j < 16; ++j) {
    int row = chunk * 16 + j;
    const float4* fp = (const float4*)(l2_pts + ((size_t)b * 128 + row) * 256);
    float4 q0 = fp[lane];
    float4 q1 = fp[lane + 32];
    float* d0 = XA + j * 260 + 3 + lane * 4;
    d0[0] = q0.x; d0[1] = q0.y; d0[2] = q0.z; d0[3] = q0.w;
    float* d1 = d0 + 128;
    d1[0] = q1.x; d1[1] = q1.y; d1[2] = q1.z; d1[3] = q1.w;
    if (lane < 3) XA[j * 260 + lane] = l2_xyz[((size_t)b * 128 + row) * 3 + lane];
    if (lane == 3) XA[j * 260 + 259] = 0.0f;
  }
  __syncthreads();
  layer_lds<260, 256, 16>(XA, XB, wt1, b1, lane);
  __syncthreads();
  layer_lds<256, 512, 16>(XB, XA, wt2, b2, lane);  // XA reused as 16x512
  __syncthreads();
  layer_max<512, 1024, 16>(XA, wt3, b3, lane, xout + (size_t)b * 1024, true);
}

// ---------------------------------------------------------------------------
// nf = einsum('os,bsc->boc', cr_w(64,512), node_fea(B,512,64)) + cr_b
// B from pair-packed node_fea: one b64 per fragment.
// ---------------------------------------------------------------------------
__global__ __launch_bounds__(32) void cr_kernel(
    const float* __restrict__ cr_w, const float* __restrict__ cr_b,
    const float* __restrict__ nfp,  // [B][256][64][2] pair-packed
    float* __restrict__ nfout) {    // [B][64][64]
  int blk = blockIdx.x;
  int b = blk >> 2;
  int m0 = (blk & 3) * 16;
  int lane = threadIdx.x;
  int half = lane >> 4, l = lane & 15;
  const v2f* nfb = (const v2f*)(nfp + ((size_t)b << 15));
  for (int n0 = 0; n0 < 64; n0 += 16) {
    v8f acc = {};
#pragma unroll 8
    for (int k0 = 0; k0 < 512; k0 += 4) {
      v2f a = *(const v2f*)(cr_w + (size_t)(m0 + l) * 512 + k0 + half * 2);
      v2f bb = nfb[(size_t)((k0 >> 1) + half) * 64 + n0 + l];
      acc = wmma4(a, bb, acc);
    }
    int rb = m0 + (half << 3);
#pragma unroll
    for (int v = 0; v < 8; ++v)
      nfout[((size_t)b * 64 + rb + v) * 64 + (n0 + l)] = acc[v] + cr_b[rb + v];
  }
}

// ---------------------------------------------------------------------------
// Host launcher
// ---------------------------------------------------------------------------
extern "C" void kernel_launch(void* const* d_in, const int* in_sizes, int n_in,
                              void* d_out, int out_size, void* d_ws,
                              size_t ws_size, hipStream_t stream) {
  (void)in_sizes; (void)n_in; (void)out_size; (void)ws_size;
  const float* xyz  = (const float*)d_in[0];   // [32][3][2048][1]
  const float* s1w0 = (const float*)d_in[1];   // (64,3)
  const float* s1w1 = (const float*)d_in[2];   // (64,64)
  const float* s1w2 = (const float*)d_in[3];   // (128,64)
  const float* s1b0 = (const float*)d_in[4];
  const float* s1b1 = (const float*)d_in[5];
  const float* s1b2 = (const float*)d_in[6];
  const float* s2w0 = (const float*)d_in[7];   // (128,131)
  const float* s2w1 = (const float*)d_in[8];   // (128,128)
  const float* s2w2 = (const float*)d_in[9];   // (256,128)
  const float* s2b0 = (const float*)d_in[10];
  const float* s2b1 = (const float*)d_in[11];
  const float* s2b2 = (const float*)d_in[12];
  const float* s3w0 = (const float*)d_in[13];  // (256,259)
  const float* s3w1 = (const float*)d_in[14];  // (512,256)
  const float* s3w2 = (const float*)d_in[15];  // (1024,512)
  const float* s3b0 = (const float*)d_in[16];
  const float* s3b1 = (const float*)d_in[17];
  const float* s3b2 = (const float*)d_in[18];
  const float* crw  = (const float*)d_in[19];  // (64,512)
  const float* crb  = (const float*)d_in[20];  // (64,)

  float* xout  = (float*)d_out;     // [32][1024]
  float* nfout = xout + 32 * 1024;  // [32][64][64][1]

  char* ws = (char*)d_ws;
  size_t off = 0;
  auto carve = [&](size_t bytes) {
    char* p = ws + off;
    off = (off + bytes + 255) & ~(size_t)255;
    return p;
  };
  float* pts   = (float*)carve((size_t)32 * 2048 * 3 * 4);
  int*   fidx1 = (int*)  carve((size_t)32 * 512 * 4);
  float* nxz1  = (float*)carve((size_t)32 * 512 * 3 * 4);   // l1_xyz
  int*   gidx1 = (int*)  carve((size_t)32 * 512 * 32 * 4);
  float* l1p   = (float*)carve((size_t)32 * 512 * 128 * 4); // l1_pts
  float* nodef = (float*)carve((size_t)32 * 512 * 64 * 4);  // node_fea
  float* nfp   = (float*)carve((size_t)32 * 512 * 64 * 4);  // pair-packed
  int*   fidx2 = (int*)  carve((size_t)32 * 128 * 4);
  float* nxz2  = (float*)carve((size_t)32 * 128 * 3 * 4);   // l2_xyz
  int*   gidx2 = (int*)  carve((size_t)32 * 128 * 64 * 4);
  float* l2p   = (float*)carve((size_t)32 * 128 * 256 * 4); // l2_pts
  // pair-packed, zero-padded transposed weights
  float* wt1a = (float*)carve((size_t)4 * 64 * 4);
  float* wt1b = (float*)carve((size_t)64 * 64 * 4);
  float* wt1c = (float*)carve((size_t)64 * 128 * 4);
  float* wt2a = (float*)carve((size_t)132 * 128 * 4);
  float* wt2b = (float*)carve((size_t)128 * 128 * 4);
  float* wt2c = (float*)carve((size_t)128 * 256 * 4);
  float* wt3a = (float*)carve((size_t)260 * 256 * 4);
  float* wt3b = (float*)carve((size_t)256 * 512 * 4);
  float* wt3c = (float*)carve((size_t)512 * 1024 * 4);

  auto wtr = [&](const float* w, float* wtp, int kreal, int kpad, int cout) {
    int n = kpad * cout;
    wtr_kernel<<<(n + 255) / 256, 256, 0, stream>>>(w, wtp, kreal, kpad, cout);
  };
  wtr(s1w0, wt1a, 3, 4, 64);
  wtr(s1w1, wt1b, 64, 64, 64);
  wtr(s1w2, wt1c, 64, 64, 128);
  wtr(s2w0, wt2a, 131, 132, 128);
  wtr(s2w1, wt2b, 128, 128, 128);
  wtr(s2w2, wt2c, 128, 128, 256);
  wtr(s3w0, wt3a, 259, 260, 256);
  wtr(s3w1, wt3b, 256, 256, 512);
  wtr(s3w2, wt3c, 512, 512, 1024);

  // 0) layout + zero-init
  transpose_kernel<<<dim3(8, 32), 256, 0, stream>>>(xyz, pts);
  zero_kernel<<<4096, 256, 0, stream>>>(xout, l2p);

  // 1) SA1
  fps_kernel<<<32, 256, 0, stream>>>(pts, 2048, 512, fidx1);
  gather_xyz_kernel<<<dim3(2, 32), 256, 0, stream>>>(pts, fidx1, nxz1, 2048, 512);
  ball_query_kernel<<<2048, 256, 0, stream>>>(pts, nxz1, 2048, 512, 32,
                                              0.2f * 0.2f, 32 * 512, gidx1);
  sa1_kernel<<<32 * 512, 32, 0, stream>>>(pts, nxz1, gidx1, wt1a, s1b0, wt1b,
                                          s1b1, wt1c, s1b2, l1p, nodef);

  // 2) SA2
  fps_kernel<<<32, 256, 0, stream>>>(nxz1, 512, 128, fidx2);
  gather_xyz_kernel<<<dim3(1, 32), 256, 0, stream>>>(nxz1, fidx2, nxz2, 512, 128);
  ball_query_kernel<<<512, 256, 0, stream>>>(nxz1, nxz2, 512, 128, 64,
                                             0.4f * 0.4f, 32 * 128, gidx2);
  sa2_kernel<<<32 * 128 * 2, 32, 0, stream>>>(nxz1, l1p, nxz2, gidx2, wt2a,
                                              s2b0, wt2b, s2b1, wt2c, s2b2, l2p);

  // 3) SA3 (group-all) -> d_out[:32768]
  sa3_kernel<<<32 * 8, 32, 0, stream>>>(nxz2, l2p, wt3a, s3b0, wt3b, s3b1,
                                        wt3c, s3b2, xout);

  // 4) node-feature projection -> d_out[32768:]
  nfp_kernel<<<4096, 256, 0, stream>>>(nodef, nfp);
  cr_kernel<<<32 * 4, 32, 0, stream>>>(crw, crb, nfp, nfout);
}